// CustomNoisyTopExpertsPerItemRouter_24223615549769
// MI455X (gfx1250) — compile-verified
//
#include <hip/hip_runtime.h>
#include <hip/hip_bf16.h>
#include <math.h>

// ---------------------------------------------------------------------------
// Noisy top-2 MoE router for MI455X (gfx1250, wave32).
//   G=16 groups, S=1024 tokens, D=4096 features, E=64 experts, K=2, C=64.
// Outputs (flat, in order): combine[G,S,E,C], aux[G], imp[G], load[G],
//                           gates_noisy[G,S,E]
// ---------------------------------------------------------------------------

#define G_ 16
#define S_ 1024
#define D_ 4096
#define E_ 64
#define CAP_ 64
#define TOK_ (G_ * S_)

typedef __attribute__((ext_vector_type(2))) float v2f;
typedef __attribute__((ext_vector_type(4))) float f4;
typedef __attribute__((ext_vector_type(8))) float v8f;

// ---------------------------------------------------------------------------
// Kernel 1: logits = inputs @ W via V_WMMA_F32_16X16X4_F32.
// grid.x = TOK_/16 (M tiles), 128 threads = 4 waves, wave w owns N-tile w.
// A 16x4 fp32 layout: lanes 0-15 hold M=lane, VGPR0=K0,VGPR1=K1;
//                     lanes 16-31 hold M=lane-16, VGPR0=K2,VGPR1=K3.
// B 4x16  fp32 layout: VGPR0 lanes 0-15 = row K0, lanes 16-31 = row K2;
//                      VGPR1 lanes 0-15 = row K1, lanes 16-31 = row K3.
// C/D 16x16 fp32: VGPR v: lanes 0-15 -> M=v, lanes 16-31 -> M=v+8, N=lane&15.
// ---------------------------------------------------------------------------
__global__ __launch_bounds__(128) void gemm_logits_wmma(
    const float* __restrict__ X, const float* __restrict__ W,
    float* __restrict__ logits) {
  const int mtile = blockIdx.x;           // 0..TOK_/16-1
  const int ntile = threadIdx.x >> 5;     // 0..3
  const int lane  = threadIdx.x & 31;
  const int row   = lane & 15;
  const int half  = lane >> 4;            // 0 or 1

  const float* Arow = X + (size_t)(mtile * 16 + row) * D_ + 2 * half;
  const float* Bcol = W + (size_t)(2 * half) * E_ + ntile * 16 + row;

  v8f acc = {};
  for (int kk = 0; kk < D_; kk += 128) {
    __builtin_prefetch(Arow + kk + 128, 0, 0);   // global_prefetch_b8 stream hint
#pragma unroll 4
    for (int k0 = kk; k0 < kk + 128; k0 += 4) {
      v2f a;
      a.x = Arow[k0];
      a.y = Arow[k0 + 1];
      v2f b;
      b.x = Bcol[(size_t)k0 * E_];
      b.y = Bcol[(size_t)k0 * E_ + E_];
      acc = __builtin_amdgcn_wmma_f32_16x16x4_f32(
          /*neg_a=*/false, a, /*neg_b=*/false, b,
          /*c_mod=*/(short)0, acc, /*reuse_a=*/false, /*reuse_b=*/false);
    }
  }

  float* out = logits + (size_t)(mtile * 16) * E_ + ntile * 16 + row;
#pragma unroll
  for (int v = 0; v < 8; ++v) {
    const int M = v + 8 * half;
    out[(size_t)M * E_] = acc[v];
  }
}

// ---------------------------------------------------------------------------
// Kernel 2: per-token routing math. One wave32 per token; lane owns experts
// lane and lane+32.
// ---------------------------------------------------------------------------
__global__ __launch_bounds__(256) void router_token(
    const float* __restrict__ logits, const float* __restrict__ noise,
    float* __restrict__ gates_ws, float* __restrict__ p_ws,
    int* __restrict__ top_e_ws, float* __restrict__ top_g_ws,
    float* __restrict__ gates_noisy_out) {
  const int token = blockIdx.x * 8 + (threadIdx.x >> 5);  // g*S + s
  const int lane  = threadIdx.x & 31;
  const float NSTD     = 1.0f / 64.0f;    // NOISE_STD / NUM_EXPERTS
  const float INV_NSTD = 64.0f;
  const float INV_SQRT2 = 0.70710678118654752f;

  const size_t base = (size_t)token * E_;
  const float l0 = logits[base + lane];
  const float l1 = logits[base + lane + 32];
  const float n0 = l0 + NSTD * noise[base + lane];
  const float n1 = l1 + NSTD * noise[base + lane + 32];

  // --- softmax(logits) -> gates (workspace) ---
  float m = fmaxf(l0, l1);
#pragma unroll
  for (int o = 16; o; o >>= 1) m = fmaxf(m, __shfl_xor(m, o, 32));
  const float e0 = __expf(l0 - m), e1 = __expf(l1 - m);
  float s = e0 + e1;
#pragma unroll
  for (int o = 16; o; o >>= 1) s += __shfl_xor(s, o, 32);
  const float inv = 1.0f / s;
  gates_ws[base + lane]      = e0 * inv;
  gates_ws[base + lane + 32] = e1 * inv;

  // --- softmax(noisy logits) -> gates_noisy (output) ---
  float mn = fmaxf(n0, n1);
#pragma unroll
  for (int o = 16; o; o >>= 1) mn = fmaxf(mn, __shfl_xor(mn, o, 32));
  const float f0 = __expf(n0 - mn), f1 = __expf(n1 - mn);
  float sn = f0 + f1;
#pragma unroll
  for (int o = 16; o; o >>= 1) sn += __shfl_xor(sn, o, 32);
  const float invn = 1.0f / sn;
  gates_noisy_out[base + lane]      = f0 * invn;
  gates_noisy_out[base + lane + 32] = f1 * invn;

  // --- wave-wide top-2 of noisy logits (tie-break: lower index wins) ---
  float a, bv;
  int ia, ib;
  if (n0 >= n1) { a = n0; ia = lane;      bv = n1; ib = lane + 32; }
  else          { a = n1; ia = lane + 32; bv = n0; ib = lane; }
#pragma unroll
  for (int o = 16; o; o >>= 1) {
    const float oa = __shfl_xor(a, o, 32);
    const int   oia = __shfl_xor(ia, o, 32);
    const float ob = __shfl_xor(bv, o, 32);
    const int   oib = __shfl_xor(ib, o, 32);
    const bool first = (a > oa) || (a == oa && ia < oia);
    float na, nb;
    int nia, nib;
    if (first) {
      na = a; nia = ia;
      const bool sec = (bv > oa) || (bv == oa && ib < oia);
      nb = sec ? bv : oa; nib = sec ? ib : oia;
    } else {
      na = oa; nia = oia;
      const bool sec = (ob > a) || (ob == a && oib < ia);
      nb = sec ? ob : a; nib = sec ? oib : ia;
    }
    a = na; ia = nia; bv = nb; ib = nib;
  }
  const float thr = bv;  // 2nd-largest noisy logit
  if (lane == 0) {
    top_e_ws[token * 2]     = ia;
    top_e_ws[token * 2 + 1] = ib;
    top_g_ws[token * 2]     = __expf(a - mn) * invn;   // gates_noisy at top-1
    top_g_ws[token * 2 + 1] = __expf(bv - mn) * invn;  // gates_noisy at top-2
  }

  // --- p = 1 - Phi((thr - logit)/nstd) = 0.5*erfc((thr-logit)/(nstd*sqrt2))
  p_ws[base + lane]      = 0.5f * erfcf((thr - l0) * INV_NSTD * INV_SQRT2);
  p_ws[base + lane + 32] = 0.5f * erfcf((thr - l1) * INV_NSTD * INV_SQRT2);
}

// ---------------------------------------------------------------------------
// Kernel 3: buffer positions. Priority order = k-major (all 1st choices in
// token order, then all 2nd choices). Each thread owns one expert and counts.
// grid.x = G, 64 threads.
// ---------------------------------------------------------------------------
__global__ __launch_bounds__(64) void positions(
    const int* __restrict__ top_e_ws, int* __restrict__ pos_ws) {
  __shared__ short eid[2 * S_];
  const int g = blockIdx.x;
  const int t = threadIdx.x;
  for (int i = t; i < 2 * S_; i += 64) {
    const int k = i >> 10, s2 = i & (S_ - 1);
    eid[i] = (short)top_e_ws[((size_t)g * S_ + s2) * 2 + k];
  }
  __syncthreads();
  int cnt = 0;
  for (int i = 0; i < 2 * S_; ++i) {
    if (eid[i] == t) pos_ws[(size_t)g * 2 * S_ + i] = cnt++;
  }
}

// ---------------------------------------------------------------------------
// Kernel 4: combine[G,S,E,C] streaming fill (268 MB). One block per token,
// branchless select vs the <=2 scatter targets, float4 nontemporal stores.
// ---------------------------------------------------------------------------
__global__ __launch_bounds__(256) void combine_fill(
    const int* __restrict__ top_e_ws, const float* __restrict__ top_g_ws,
    const int* __restrict__ pos_ws, float* __restrict__ out) {
  const int token = blockIdx.x;            // g*S + s
  const int g = token >> 10, s2 = token & (S_ - 1);
  const int e1 = top_e_ws[token * 2];
  const int e2 = top_e_ws[token * 2 + 1];
  const float g1 = top_g_ws[token * 2];
  const float g2 = top_g_ws[token * 2 + 1];
  const int p1 = pos_ws[(size_t)g * 2 * S_ + s2];        // k = 0 entry
  const int p2 = pos_ws[(size_t)g * 2 * S_ + S_ + s2];   // k = 1 entry
  const int tgt1 = (p1 < CAP_) ? e1 * CAP_ + p1 : -1;
  const int tgt2 = (p2 < CAP_) ? e2 * CAP_ + p2 : -1;

  f4* dst = (f4*)(out + (size_t)token * (E_ * CAP_));
#pragma unroll
  for (int it = 0; it < 4; ++it) {
    const int vidx = it * 256 + threadIdx.x;  // 0..1023 float4 slots
    const int b = vidx * 4;
    f4 v;
    v.x = (b + 0 == tgt1) ? g1 : (b + 0 == tgt2) ? g2 : 0.0f;
    v.y = (b + 1 == tgt1) ? g1 : (b + 1 == tgt2) ? g2 : 0.0f;
    v.z = (b + 2 == tgt1) ? g1 : (b + 2 == tgt2) ? g2 : 0.0f;
    v.w = (b + 3 == tgt1) ? g1 : (b + 3 == tgt2) ? g2 : 0.0f;
    __builtin_nontemporal_store(v, dst + vidx);
  }
}

// ---------------------------------------------------------------------------
// Kernel 5: per-group losses. (std/mean)^2 is scale-invariant so sums over S
// suffice (no /S). grid.x = G, 64 threads (one per expert).
// ---------------------------------------------------------------------------
__global__ __launch_bounds__(64) void losses(
    const float* __restrict__ gates_ws, const float* __restrict__ p_ws,
    float* __restrict__ aux_out, float* __restrict__ imp_out,
    float* __restrict__ load_out) {
  const int g = blockIdx.x, e = threadIdx.x;
  const float* gb = gates_ws + (size_t)g * S_ * E_ + e;
  const float* pb = p_ws + (size_t)g * S_ * E_ + e;
  float si = 0.0f, sp = 0.0f;
  for (int s2 = 0; s2 < S_; ++s2) {
    si += gb[(size_t)s2 * E_];   // coalesced across the 64 threads
    sp += pb[(size_t)s2 * E_];
  }
  __shared__ float shi[E_];
  __shared__ float shp[E_];
  shi[e] = si;
  shp[e] = sp;
  __syncthreads();
  if (e == 0) {
    float s1 = 0.f, q1 = 0.f, s2s = 0.f, q2 = 0.f;
    for (int i = 0; i < E_; ++i) {
      s1 += shi[i]; q1 += shi[i] * shi[i];
      s2s += shp[i]; q2 += shp[i] * shp[i];
    }
    const float m1 = s1 / E_, m2 = s2s / E_;
    const float v1 = q1 / E_ - m1 * m1;   // population variance (ddof=0)
    const float v2 = q2 / E_ - m2 * m2;
    const float li = v1 / (m1 * m1);      // (std/mean)^2
    const float ll = v2 / (m2 * m2);
    imp_out[g]  = li;
    load_out[g] = ll;
    aux_out[g]  = li + ll;                // IMP_W = LOAD_W = 1
  }
}

// ---------------------------------------------------------------------------
extern "C" void kernel_launch(void* const* d_in, const int* in_sizes, int n_in,
                              void* d_out, int out_size, void* d_ws,
                              size_t ws_size, hipStream_t stream) {
  const float* inputs = (const float*)d_in[0];  // [G,S,D]
  const float* W      = (const float*)d_in[1];  // [D,E]
  const float* noise  = (const float*)d_in[2];  // [G,S,E]

  // Workspace carve-up (floats/ints).
  float* logits = (float*)d_ws;                     // TOK_*E_      = 1 Mi f32
  float* gates  = logits + (size_t)TOK_ * E_;       // TOK_*E_
  float* p_ws   = gates + (size_t)TOK_ * E_;        // TOK_*E_
  float* top_g  = p_ws + (size_t)TOK_ * E_;         // TOK_*2
  int*   top_e  = (int*)(top_g + (size_t)TOK_ * 2); // TOK_*2
  int*   pos    = top_e + (size_t)TOK_ * 2;         // TOK_*2

  // Output carve-up (flat, in reference return order).
  float* out        = (float*)d_out;
  float* combine    = out;                                     // G*S*E*C
  float* aux_out    = out + (size_t)TOK_ * E_ * CAP_;          // G
  float* imp_out    = aux_out + G_;                            // G
  float* load_out   = imp_out + G_;                            // G
  float* gatesnoisy = load_out + G_;                           // G*S*E

  gemm_logits_wmma<<<TOK_ / 16, 128, 0, stream>>>(inputs, W, logits);
  router_token<<<TOK_ / 8, 256, 0, stream>>>(logits, noise, gates, p_ws,
                                             top_e, top_g, gatesnoisy);
  positions<<<G_, 64, 0, stream>>>(top_e, pos);
  combine_fill<<<TOK_, 256, 0, stream>>>(top_e, top_g, pos, combine);
  losses<<<G_, 64, 0, stream>>>(gates, p_ws, aux_out, imp_out, load_out);
}